// GNN_40450001994248
// MI455X (gfx1250) — compile-verified
//
#include <hip/hip_runtime.h>
#include <hip/hip_bf16.h>

#define N_NODES 50000
#define N_EDGES 800000
#define DIM     128
#define DIM2    256
#define NLAYERS 5
#define NGRAPHS 256
#define NCLASS  10
#define NIN     16
#define BN_EPS  1e-5f

typedef __attribute__((ext_vector_type(2))) float v2f;
typedef __attribute__((ext_vector_type(8))) float v8f;

#if defined(__has_builtin)
#if __has_builtin(__builtin_amdgcn_wmma_f32_16x16x4_f32)
#define HAVE_WMMA_F32 1
#endif
#endif

// ---------------------------------------------------------------------------
// Node encoder: h[n,d] = sum_k x[n,k] * W_enc[k,d] + b_enc[d]
// grid = N_NODES blocks, block = 128 threads (one per d)
// ---------------------------------------------------------------------------
__global__ void gnn_encoder(const float* __restrict__ x,
                            const float* __restrict__ Wenc,
                            const float* __restrict__ benc,
                            float* __restrict__ h) {
    __shared__ float xr[NIN];
    int n = blockIdx.x;
    int d = threadIdx.x;
    if (threadIdx.x < NIN) xr[threadIdx.x] = x[(size_t)n * NIN + threadIdx.x];
    __syncthreads();
    float acc = benc[d];
#pragma unroll
    for (int k = 0; k < NIN; ++k) acc = fmaf(xr[k], Wenc[k * DIM + d], acc);
    h[(size_t)n * DIM + d] = acc;
}

// ---------------------------------------------------------------------------
// Message + aggregate: agg[dst] += relu(h[src] + edge_attr)
// agg pre-initialized with a copy of h (implements h + sum(msg)).
// One thread per (edge, 4-wide feature chunk): E * 32 threads.
// h (25.6 MB) is L2-resident; edge_attr streams from HBM once per layer.
// ---------------------------------------------------------------------------
__global__ void gnn_msg_agg(const float* __restrict__ h,
                            const float* __restrict__ ea,
                            const int*   __restrict__ src,
                            const int*   __restrict__ dst,
                            float* __restrict__ agg) {
    long long idx = (long long)blockIdx.x * blockDim.x + threadIdx.x;
    const int D4 = DIM / 4;
    long long e = idx / D4;
    int d4 = (int)(idx % D4);
    if (e >= N_EDGES) return;
    int s = src[e];
    int t = dst[e];
    const float4 hv = ((const float4*)(h + (size_t)s * DIM))[d4];
    const float4 ev = ((const float4*)(ea + (size_t)e * DIM))[d4];
    float4 m;
    m.x = fmaxf(hv.x + ev.x, 0.f);
    m.y = fmaxf(hv.y + ev.y, 0.f);
    m.z = fmaxf(hv.z + ev.z, 0.f);
    m.w = fmaxf(hv.w + ev.w, 0.f);
    float* ap = agg + (size_t)t * DIM + d4 * 4;
    atomicAdd(ap + 0, m.x);
    atomicAdd(ap + 1, m.y);
    atomicAdd(ap + 2, m.z);
    atomicAdd(ap + 3, m.w);
}

// ---------------------------------------------------------------------------
// WMMA f32 GEMM: out[rows,KOUT] = X[rows,K] @ W[K,KOUT] + bias
// grid = rows/16 blocks, block = 128 threads (4 waves).
// One block computes a FULL 16-row x KOUT output stripe:
//   wave w owns tiles at columns { t*64 + w*16 : t in [0, KOUT/64) }.
// A-tile (16 x K, contiguous in X) staged once in LDS via float4 copies;
// each A-fragment is reused across KOUT/64 WMMAs (independent accum chains
// hide the WMMA->WMMA RAW hazard). W is L2-resident (<=128 KB).
// ---------------------------------------------------------------------------
template <int K, int KOUT>
__global__ void gnn_gemm(const float* __restrict__ X,
                         const float* __restrict__ W,
                         const float* __restrict__ bias,
                         float* __restrict__ out) {
    constexpr int NT = KOUT / 64;     // column tiles per wave
    __shared__ float lx[16 * K];      // 8 KB (K=128) or 16 KB (K=256)
    const int row0 = blockIdx.x * 16;

    // contiguous 16xK tile copy, fully coalesced 128-bit loads
    {
        const float4* xs = (const float4*)(X + (size_t)row0 * K);
        float4* ls = (float4*)lx;
#pragma unroll
        for (int i = threadIdx.x; i < 16 * K / 4; i += 128) ls[i] = xs[i];
    }
    __syncthreads();

    const int wave = threadIdx.x / 32;
    const int lane = threadIdx.x % 32;

#if HAVE_WMMA_F32
    v8f acc[NT];
#pragma unroll
    for (int t = 0; t < NT; ++t) acc[t] = (v8f){};

    const int m  = lane % 16;                // A row / B col index
    const int kk = (lane / 16) * 2;          // K sub-offset for this half-wave
#pragma unroll 4
    for (int k0 = 0; k0 < K; k0 += 4) {
        v2f a;
        a[0] = lx[m * K + k0 + kk];
        a[1] = lx[m * K + k0 + kk + 1];
        const float* wrow = W + (size_t)(k0 + kk) * KOUT + wave * 16 + m;
#pragma unroll
        for (int t = 0; t < NT; ++t) {
            v2f b;
            b[0] = wrow[t * 64];
            b[1] = wrow[t * 64 + KOUT];
            acc[t] = __builtin_amdgcn_wmma_f32_16x16x4_f32(
                         /*neg_a=*/false, a, /*neg_b=*/false, b,
                         /*c_mod=*/(short)0, acc[t],
                         /*reuse_a=*/false, /*reuse_b=*/false);
        }
    }
    // D layout: reg r -> M = r + 8*(lane/16), N = lane%16
    const int mhalf = (lane / 16) * 8;
#pragma unroll
    for (int t = 0; t < NT; ++t) {
        const int nIdx = t * 64 + wave * 16 + m;
        const float bv = bias[nIdx];
#pragma unroll
        for (int r = 0; r < 8; ++r)
            out[(size_t)(row0 + r + mhalf) * KOUT + nIdx] = acc[t][r] + bv;
    }
#else
    // scalar fallback (should not be taken on gfx1250)
    for (int oi = threadIdx.x; oi < 16 * KOUT; oi += 128) {
        int r = oi / KOUT, nn = oi % KOUT;
        float a2 = bias[nn];
        for (int k = 0; k < K; ++k)
            a2 = fmaf(lx[r * K + k], W[(size_t)k * KOUT + nn], a2);
        out[(size_t)(row0 + r) * KOUT + nn] = a2;
    }
    (void)wave; (void)lane;
#endif
}

// ---------------------------------------------------------------------------
// BatchNorm pass 1: per-column partial sums (coalesced), per-block atomics.
// ---------------------------------------------------------------------------
__global__ void gnn_bn_partial(const float* __restrict__ z, int rows, int F,
                               float* __restrict__ sum, float* __restrict__ sumsq,
                               int rowsPerBlock) {
    int r0 = blockIdx.x * rowsPerBlock;
    int f = threadIdx.x % F;
    int rOff = threadIdx.x / F;
    int stride = blockDim.x / F;
    int rEnd = r0 + rowsPerBlock;
    if (rEnd > rows) rEnd = rows;
    float s = 0.f, s2 = 0.f;
    for (int r = r0 + rOff; r < rEnd; r += stride) {
        float v = z[(size_t)r * F + f];
        s += v;
        s2 += v * v;
    }
    atomicAdd(&sum[f], s);
    atomicAdd(&sumsq[f], s2);
}

__global__ void gnn_bn_finalize(const float* __restrict__ sum,
                                const float* __restrict__ sumsq,
                                int rows, int F,
                                float* __restrict__ mean,
                                float* __restrict__ istd) {
    int f = blockIdx.x * blockDim.x + threadIdx.x;
    if (f >= F) return;
    float m = sum[f] / (float)rows;
    float var = sumsq[f] / (float)rows - m * m;
    mean[f] = m;
    istd[f] = rsqrtf(var + BN_EPS);
}

__global__ void gnn_bn_apply(float* __restrict__ z,
                             const float* __restrict__ mean,
                             const float* __restrict__ istd,
                             const float* __restrict__ gamma,
                             const float* __restrict__ beta,
                             long long total, int F, int doRelu) {
    long long idx = (long long)blockIdx.x * blockDim.x + threadIdx.x;
    if (idx >= total) return;
    int f = (int)(idx % F);
    float v = (z[idx] - mean[f]) * istd[f] * gamma[f] + beta[f];
    z[idx] = doRelu ? fmaxf(v, 0.f) : v;
}

// ---------------------------------------------------------------------------
// Global mean pool (atomics) + output projection
// ---------------------------------------------------------------------------
__global__ void gnn_pool_sum(const float* __restrict__ h,
                             const int* __restrict__ batch,
                             float* __restrict__ gsum,
                             float* __restrict__ gcnt) {
    long long idx = (long long)blockIdx.x * blockDim.x + threadIdx.x;
    if (idx >= (long long)N_NODES * DIM) return;
    int n = (int)(idx / DIM);
    int d = (int)(idx % DIM);
    int g = batch[n];
    atomicAdd(&gsum[(size_t)g * DIM + d], h[idx]);
    if (d == 0) atomicAdd(&gcnt[g], 1.0f);
}

__global__ void gnn_out(const float* __restrict__ gsum,
                        const float* __restrict__ gcnt,
                        const float* __restrict__ Wout,
                        const float* __restrict__ bout,
                        float* __restrict__ out) {
    int idx = blockIdx.x * blockDim.x + threadIdx.x;
    if (idx >= NGRAPHS * NCLASS) return;
    int g = idx / NCLASS, c = idx % NCLASS;
    float inv = 1.0f / fmaxf(gcnt[g], 1.0f);
    float acc = bout[c];
#pragma unroll 8
    for (int d = 0; d < DIM; ++d)
        acc = fmaf(gsum[(size_t)g * DIM + d] * inv, Wout[d * NCLASS + c], acc);
    out[idx] = acc;
}

// ---------------------------------------------------------------------------
// Host-side orchestration (all work on `stream`, graph-capture safe)
// ---------------------------------------------------------------------------
extern "C" void kernel_launch(void* const* d_in, const int* in_sizes, int n_in,
                              void* d_out, int out_size, void* d_ws, size_t ws_size,
                              hipStream_t stream) {
    const float* x    = (const float*)d_in[0];
    const float* ea   = (const float*)d_in[1];
    const float* Wenc = (const float*)d_in[2];
    const float* benc = (const float*)d_in[3];
    const float* W1   = (const float*)d_in[4];   // [L, 128, 256]
    const float* b1   = (const float*)d_in[5];   // [L, 256]
    const float* g1   = (const float*)d_in[6];
    const float* be1  = (const float*)d_in[7];
    const float* W2   = (const float*)d_in[8];   // [L, 256, 128]
    const float* b2   = (const float*)d_in[9];   // [L, 128]
    const float* g_bn = (const float*)d_in[10];
    const float* b_bn = (const float*)d_in[11];
    const float* Wout = (const float*)d_in[12];
    const float* bout = (const float*)d_in[13];
    const int*   ei   = (const int*)d_in[14];    // [2, E]
    const int*   batch= (const int*)d_in[15];
    const int* src = ei;
    const int* dst = ei + N_EDGES;

    float* ws = (float*)d_ws;
    float* hA   = ws;                                   // N*128
    float* hB   = hA + (size_t)N_NODES * DIM;           // N*128
    float* agg  = hB + (size_t)N_NODES * DIM;           // N*128
    float* z    = agg + (size_t)N_NODES * DIM;          // N*256
    float* stat = z + (size_t)N_NODES * DIM2;
    float* sumb = stat;
    float* sqb  = stat + 256;
    float* mean = stat + 512;
    float* istd = stat + 768;
    float* gsum = stat + 1024;                          // G*128
    float* gcnt = gsum + (size_t)NGRAPHS * DIM;         // G

    const int ROWS_PER_BLK = 256;
    const int bnBlocks = (N_NODES + ROWS_PER_BLK - 1) / ROWS_PER_BLK;

    gnn_encoder<<<N_NODES, DIM, 0, stream>>>(x, Wenc, benc, hA);

    float* hCur = hA;
    float* hNext = hB;
    for (int l = 0; l < NLAYERS; ++l) {
        hipMemcpyAsync(agg, hCur, (size_t)N_NODES * DIM * sizeof(float),
                       hipMemcpyDeviceToDevice, stream);
        {
            long long th = (long long)N_EDGES * (DIM / 4);
            gnn_msg_agg<<<(unsigned)((th + 255) / 256), 256, 0, stream>>>(
                hCur, ea, src, dst, agg);
        }
        gnn_gemm<DIM, DIM2><<<N_NODES / 16, 128, 0, stream>>>(
            agg, W1 + (size_t)l * DIM * DIM2, b1 + (size_t)l * DIM2, z);
        hipMemsetAsync(sumb, 0, 512 * sizeof(float), stream);
        gnn_bn_partial<<<bnBlocks, 256, 0, stream>>>(z, N_NODES, DIM2, sumb, sqb, ROWS_PER_BLK);
        gnn_bn_finalize<<<1, 256, 0, stream>>>(sumb, sqb, N_NODES, DIM2, mean, istd);
        {
            long long tot = (long long)N_NODES * DIM2;
            gnn_bn_apply<<<(unsigned)((tot + 255) / 256), 256, 0, stream>>>(
                z, mean, istd, g1 + (size_t)l * DIM2, be1 + (size_t)l * DIM2, tot, DIM2, 1);
        }
        gnn_gemm<DIM2, DIM><<<N_NODES / 16, 128, 0, stream>>>(
            z, W2 + (size_t)l * DIM2 * DIM, b2 + (size_t)l * DIM, hNext);
        hipMemsetAsync(sumb, 0, 512 * sizeof(float), stream);
        gnn_bn_partial<<<bnBlocks, 256, 0, stream>>>(hNext, N_NODES, DIM, sumb, sqb, ROWS_PER_BLK);
        gnn_bn_finalize<<<1, 256, 0, stream>>>(sumb, sqb, N_NODES, DIM, mean, istd);
        {
            long long tot = (long long)N_NODES * DIM;
            gnn_bn_apply<<<(unsigned)((tot + 255) / 256), 256, 0, stream>>>(
                hNext, mean, istd, g_bn + (size_t)l * DIM, b_bn + (size_t)l * DIM,
                tot, DIM, (l < NLAYERS - 1) ? 1 : 0);
        }
        float* t = hCur; hCur = hNext; hNext = t;
    }

    hipMemsetAsync(gsum, 0, ((size_t)NGRAPHS * DIM + NGRAPHS) * sizeof(float), stream);
    {
        long long th = (long long)N_NODES * DIM;
        gnn_pool_sum<<<(unsigned)((th + 255) / 256), 256, 0, stream>>>(hCur, batch, gsum, gcnt);
    }
    gnn_out<<<(NGRAPHS * NCLASS + 255) / 256, 256, 0, stream>>>(
        gsum, gcnt, Wout, bout, (float*)d_out);
}